// LagrangeKANNinner_4578435137545
// MI455X (gfx1250) — compile-verified
//
#include <hip/hip_runtime.h>
#include <hip/hip_bf16.h>

typedef __attribute__((ext_vector_type(2))) float v2f;
typedef __attribute__((ext_vector_type(8))) float v8f;

#define N_COLL   4096
#define KW       64
#define NNODES   257
#define T_ELEMS  (N_COLL * KW)              // 262144
#define PHI_ELEMS ((size_t)N_COLL * KW * NNODES) // 67,371,008
#define PHI_BASE ((size_t)(3 * T_ELEMS))    // 786432
#define DELTA_X  0.0078125f                 // 0.5*4*(1-0)/256
#define INV_DX   128.0f
#define INV_DX2  16384.0f

// ---------------------------------------------------------------------------
// Lagrange basis (P=5, nodes = {-1,-0.5,0,0.5,1}) : phi, dphi, ddphi
// Fully unrolled; all node arithmetic constant-folds.
// ---------------------------------------------------------------------------
__device__ __forceinline__ void lag_basis(float xr, float* phi, float* dphi, float* ddphi) {
    const float ND[5] = {-1.0f, -0.5f, 0.0f, 0.5f, 1.0f};
#pragma unroll
    for (int j = 0; j < 5; ++j) {
        // phi_j
        float p = 1.0f;
#pragma unroll
        for (int m = 0; m < 5; ++m)
            if (m != j) p *= (xr - ND[m]) * (1.0f / (ND[j] - ND[m]));
        phi[j] = p;
        // dphi_j
        float y = 0.0f;
#pragma unroll
        for (int i = 0; i < 5; ++i) {
            if (i == j) continue;
            float k = 1.0f / (ND[j] - ND[i]);
#pragma unroll
            for (int m = 0; m < 5; ++m)
                if (m != i && m != j) k *= (xr - ND[m]) * (1.0f / (ND[j] - ND[m]));
            y += k;
        }
        dphi[j] = y;
        // ddphi_j
        float y2 = 0.0f;
#pragma unroll
        for (int i = 0; i < 5; ++i) {
            if (i == j) continue;
            float s = 0.0f;
#pragma unroll
            for (int m = 0; m < 5; ++m) {
                if (m == i || m == j) continue;
                float kp = 1.0f / (ND[j] - ND[m]);
#pragma unroll
                for (int n = 0; n < 5; ++n)
                    if (n != i && n != j && n != m)
                        kp *= (xr - ND[n]) * (1.0f / (ND[j] - ND[n]));
                s += kp;
            }
            y2 += s * (1.0f / (ND[j] - ND[i]));
        }
        ddphi[j] = y2;
    }
}

// ---------------------------------------------------------------------------
// Kernel 1: per-point basis values -> workspace; also writes delta_x scalar.
// ws layout: int nl[4096]; float vals[4096][15]  (phi*5 | dphi*5 | ddphi*5)
// ---------------------------------------------------------------------------
__global__ __launch_bounds__(256) void kan_basis(const float* __restrict__ x,
                                                 int* __restrict__ nl_arr,
                                                 float* __restrict__ vals,
                                                 float* __restrict__ dout) {
    int i = blockIdx.x * blockDim.x + threadIdx.x;
    if (i == 0) dout[PHI_BASE + 3 * PHI_ELEMS] = DELTA_X;  // last output element
    if (i >= N_COLL) return;

    float xs = 256.0f * x[i];                 // (N_NODES-1)*(x - X_MIN)/(X_MAX - X_MIN)
    int ide = (int)floorf(xs * 0.25f);
    ide = ide < 0 ? 0 : (ide > 63 ? 63 : ide);
    int nl = ide * 4;
    float xr = (xs - (float)nl - 2.0f) * 0.5f;

    float phi[5], dphi[5], ddphi[5];
    lag_basis(xr, phi, dphi, ddphi);

    nl_arr[i] = nl;
    float* v = vals + i * 15;
#pragma unroll
    for (int p = 0; p < 5; ++p) {
        v[p]      = phi[p];
        v[5 + p]  = dphi[p]  * INV_DX;
        v[10 + p] = ddphi[p] * INV_DX2;
    }
}

// ---------------------------------------------------------------------------
// Kernel 2: the 812 MB kernel. One wave per output row (3 * 4096 * 64 rows of
// 257 floats). Coalesced b32 stores; every byte written exactly once.
// ---------------------------------------------------------------------------
__global__ __launch_bounds__(256) void kan_fill(const int* __restrict__ nl_arr,
                                                const float* __restrict__ vals,
                                                float* __restrict__ dout) {
    int wave = blockIdx.x * 8 + (threadIdx.x >> 5);   // 0 .. 786431
    int lane = threadIdx.x & 31;
    int tensor = wave >> 18;          // 0,1,2  (262144 = 2^18 rows per tensor)
    int rem    = wave & 262143;       // i*64 + k
    int i      = rem >> 6;

    int nl = nl_arr[i];
    const float* v = vals + i * 15 + tensor * 5;
    float v0 = v[0], v1 = v[1], v2 = v[2], v3 = v[3], v4 = v[4];

    float* row = dout + PHI_BASE + (size_t)tensor * PHI_ELEMS + (size_t)rem * NNODES;
#pragma unroll
    for (int it = 0; it < 9; ++it) {
        int c = lane + it * 32;
        if (c < NNODES) {
            int u = c - nl;
            float val = (u == 0) ? v0 : (u == 1) ? v1 : (u == 2) ? v2
                      : (u == 3) ? v3 : (u == 4) ? v4 : 0.0f;
            row[c] = val;
        }
    }
}

// ---------------------------------------------------------------------------
// Kernel 3: t/dt/ddt via per-element binning + f32 WMMA (16x16x4).
// One block (one wave) per element e: all its points share nl = 4e, so
//   t_tile[16 pts x 16 k] = A(16x5 basis) x B(5x16 weight cols)  (f32 GEMM)
// K=5 split as K=4 WMMA + rank-1 WMMA (col/row 0 only), same C accumulator.
// ---------------------------------------------------------------------------
#if __has_builtin(__builtin_amdgcn_wmma_f32_16x16x4_f32)
#define HAVE_WMMA_F32_K4 1
#endif

__global__ __launch_bounds__(32) void kan_wmma(const float* __restrict__ W,   // 64 x 257
                                               const int* __restrict__ nl_arr,
                                               const float* __restrict__ vals,
                                               float* __restrict__ out) {    // t at 0
    __shared__ int idxs[N_COLL];
    __shared__ int cnt;
    int lane = threadIdx.x;
    int e = blockIdx.x;
    int nl = e * 4;

    if (lane == 0) cnt = 0;
    __syncthreads();
    for (int i = lane; i < N_COLL; i += 32) {
        if (nl_arr[i] == nl) { int pos = atomicAdd(&cnt, 1); idxs[pos] = i; }
    }
    __syncthreads();
    int count = cnt;
    int n  = lane & 15;
    int hi = lane >> 4;

#ifdef HAVE_WMMA_F32_K4
    // B fragments (weight only; shared by all 3 derivative levels).
    // Layout (mirrors documented f32 A 16x4): vgpr0 = K0 (lanes 0-15) / K2
    // (lanes 16-31); vgpr1 = K1 / K3.
    v2f b1[4], b2[4];
#pragma unroll
    for (int kt = 0; kt < 4; ++kt) {
        const float* wr = W + (size_t)(kt * 16 + n) * NNODES + nl;
        v2f b;
        b.x = hi ? wr[2] : wr[0];
        b.y = hi ? wr[3] : wr[1];
        b1[kt] = b;
        v2f b4;
        b4.x = hi ? 0.0f : wr[4];   // only K-row 0 carries the p=4 term
        b4.y = 0.0f;
        b2[kt] = b4;
    }

    for (int t0 = 0; t0 < count; t0 += 16) {          // uniform trip count
        bool valid = (t0 + n) < count;
        int pi = valid ? idxs[t0 + n] : 0;
        v2f A1[3], A2[3];
#pragma unroll
        for (int d = 0; d < 3; ++d) {
            const float* vv = vals + pi * 15 + d * 5;
            float a0 = hi ? vv[2] : vv[0];
            float a1 = hi ? vv[3] : vv[1];
            a0 = valid ? a0 : 0.0f;
            a1 = valid ? a1 : 0.0f;
            v2f a; a.x = a0; a.y = a1;
            A1[d] = a;
            v2f a4; a4.x = (hi || !valid) ? 0.0f : vv[4]; a4.y = 0.0f;
            A2[d] = a4;
        }
#pragma unroll
        for (int kt = 0; kt < 4; ++kt) {
#pragma unroll
            for (int d = 0; d < 3; ++d) {
                v8f c = {};
                c = __builtin_amdgcn_wmma_f32_16x16x4_f32(
                        false, A1[d], false, b1[kt], (short)0, c, false, false);
                c = __builtin_amdgcn_wmma_f32_16x16x4_f32(
                        false, A2[d], false, b2[kt], (short)0, c, false, false);
                float* od = out + d * T_ELEMS;
#pragma unroll
                for (int r = 0; r < 8; ++r) {
                    int mm = r + 8 * hi;              // C row
                    if (t0 + mm < count) {
                        int ii = idxs[t0 + mm];
                        od[(size_t)ii * KW + kt * 16 + n] = c[r];
                    }
                }
            }
        }
    }
#else
    // Fallback (no WMMA builtin): direct 5-term dot per (point, k) pair.
    for (int t = lane; t < count * KW; t += 32) {
        int ii = idxs[t >> 6];
        int k  = t & 63;
        const float* wr = W + (size_t)k * NNODES + nl;
#pragma unroll
        for (int d = 0; d < 3; ++d) {
            const float* vv = vals + ii * 15 + d * 5;
            float acc = vv[0]*wr[0] + vv[1]*wr[1] + vv[2]*wr[2] + vv[3]*wr[3] + vv[4]*wr[4];
            out[d * T_ELEMS + (size_t)ii * KW + k] = acc;
        }
    }
#endif
}

// ---------------------------------------------------------------------------
extern "C" void kernel_launch(void* const* d_in, const int* in_sizes, int n_in,
                              void* d_out, int out_size, void* d_ws, size_t ws_size,
                              hipStream_t stream) {
    const float* x      = (const float*)d_in[0];   // (4096,)
    const float* weight = (const float*)d_in[1];   // (64, 257)
    float* out = (float*)d_out;

    int*   nl_arr = (int*)d_ws;
    float* vals   = (float*)((char*)d_ws + N_COLL * sizeof(int));

    // 1) per-point basis -> workspace (+ delta_x scalar)
    kan_basis<<<N_COLL / 256, 256, 0, stream>>>(x, nl_arr, vals, out);

    // 2) 812 MB single-pass row fill of phi/dphi/ddphi_full (bandwidth bound)
    //    3 * 262144 rows, 8 waves per 256-thread block
    kan_fill<<<(3 * T_ELEMS) / 8, 256, 0, stream>>>(nl_arr, vals, out);

    // 3) t/dt/ddt via per-element binned f32 WMMA GEMM
    kan_wmma<<<64, 32, 0, stream>>>(weight, nl_arr, vals, out);
}